// QCDAwareOld_68813966017012
// MI455X (gfx1250) — compile-verified
//
#include <hip/hip_runtime.h>

typedef __attribute__((ext_vector_type(2))) float v2f;
typedef __attribute__((ext_vector_type(8))) float v8f;

#define B_DIM 8
#define F_DIM 16
#define N_PTS 2048
#define N_TILES (N_PTS / 16) // 128

// One wave32 computes one 16x16 tile of w[b] via 4x V_WMMA_F32_16X16X4_F32
// (K = 16 = 4 steps of K=4, fp32 exact like the reference).
__global__ __launch_bounds__(256) void qcd_wij_kernel(
    const float* __restrict__ emb,     // (B, F, N)
    const float* __restrict__ alpha_p, // (1,1)
    const float* __restrict__ beta_p,  // (1,1,1)
    float* __restrict__ out)           // (B, N, N)
{
    const int lane = threadIdx.x & 31;
    const int wave = threadIdx.x >> 5;
    const int jt = blockIdx.x * 8 + wave; // 0..127
    const int it = blockIdx.y;            // 0..127
    const int b  = blockIdx.z;            // 0..7

    const int i0 = it * 16;
    const int j0 = jt * 16;

    const float alpha = alpha_p[0];
    const float beta  = beta_p[0];
    const float beta2 = beta * beta;

    const float* __restrict__ E = emb + (size_t)b * F_DIM * N_PTS;

    // A-frag (16x4 f32): lane holds M = lane&15; K pair = (lane>>4)*2.
    // B-frag (4x16 f32): lane holds N = lane&15; same K split. Both operands
    // index the same axis of emb, so the loads are symmetric.
    const int m     = lane & 15;
    const int khalf = (lane >> 4) * 2; // 0 (lanes 0-15) or 2 (lanes 16-31)

    v8f c = {}; // 16x16 f32 accumulator
    float ssa = 0.f, ssb = 0.f;       // per-lane partial sum of squares
    float mom_i = 1.f, mom_j = 1.f;   // emb[b,4,*], captured at t=1, khalf==0

#pragma unroll
    for (int t = 0; t < 4; ++t) {
        const int k0 = t * 4 + khalf;
        const float a0 = E[(size_t)(k0    ) * N_PTS + i0 + m];
        const float a1 = E[(size_t)(k0 + 1) * N_PTS + i0 + m];
        const float b0 = E[(size_t)(k0    ) * N_PTS + j0 + m];
        const float b1 = E[(size_t)(k0 + 1) * N_PTS + j0 + m];

        ssa += a0 * a0 + a1 * a1;
        ssb += b0 * b0 + b1 * b1;
        if (t == 1) { // k0 == 4 on the low half-wave -> momentum row
            mom_i = (khalf == 0) ? a0 : mom_i;
            mom_j = (khalf == 0) ? b0 : mom_j;
        }

        v2f av; av[0] = a0; av[1] = a1;
        v2f bv; bv[0] = b0; bv[1] = b1;
        // 8 args: (neg_a, A, neg_b, B, c_mod, C, reuse_a, reuse_b)
        c = __builtin_amdgcn_wmma_f32_16x16x4_f32(
                false, av, false, bv, (short)0, c, false, false);
    }

    // Each half-wave accumulated half of the K range: combine across L <-> L^16.
    ssa += __shfl_xor(ssa, 16, 32);
    ssb += __shfl_xor(ssb, 16, 32);
    // Momentum only valid on lanes 0-15; broadcast to the high half.
    mom_i = __shfl(mom_i, m, 32);
    mom_j = __shfl(mom_j, m, 32);

    // pow_momenta = exp(2*alpha*log(momentum)) (momentum in (0.001, 1))
    const float pm_i = expf(2.0f * alpha * logf(mom_i));
    const float pm_j = expf(2.0f * alpha * logf(mom_j));

    // C/D layout: VGPR r -> M = r + 8*(lane>=16), N = lane&15.
    const int hi = lane >> 4;
    float* __restrict__ outp =
        out + ((size_t)b * N_PTS + (size_t)i0) * N_PTS + j0;

#pragma unroll
    for (int r = 0; r < 8; ++r) {
        const int row = r + hi * 8; // M within tile (0..15)
        // Row-indexed quantities live in lane `row` (bpermute).
        const float sq_r = __shfl(ssa,  row, 32);
        const float pm_r = __shfl(pm_i, row, 32);

        const float sqdist = sq_r + ssb - 2.0f * c[r];
        const float mn     = fminf(pm_r, pm_j);
        const float w      = expf(-beta2 * sqdist * mn);

        outp[(size_t)row * N_PTS + m] = w; // coalesced across the wave
    }
}

extern "C" void kernel_launch(void* const* d_in, const int* in_sizes, int n_in,
                              void* d_out, int out_size, void* d_ws, size_t ws_size,
                              hipStream_t stream) {
    (void)in_sizes; (void)n_in; (void)out_size; (void)d_ws; (void)ws_size;
    const float* emb     = (const float*)d_in[0];
    const float* alpha_p = (const float*)d_in[1];
    const float* beta_p  = (const float*)d_in[2];
    float* out = (float*)d_out;

    // 8 waves per block, each wave one 16x16 tile; adjacent waves -> adjacent
    // j-tiles for contiguous stores. Grid: (128/8 j-groups, 128 i-tiles, 8 b).
    dim3 grid(N_TILES / 8, N_TILES, B_DIM);
    qcd_wij_kernel<<<grid, 256, 0, stream>>>(emb, alpha_p, beta_p, out);
}